// BeliveMapsNMS_23751169147303
// MI455X (gfx1250) — compile-verified
//
#include <hip/hip_runtime.h>
#include <hip/hip_bf16.h>

#ifndef __has_builtin
#define __has_builtin(x) 0
#endif

#define NMS_H 1024
#define NMS_W 1024
#define NMS_RB 8          // output rows per phase-1 block
#define NMS_MAXROWS 14    // RB + 2*3 halo
#define NMS_K 100
#define NMS_S 25

typedef unsigned int v4u __attribute__((ext_vector_type(4)));
typedef int          v8i __attribute__((ext_vector_type(8)));
typedef int          v4i __attribute__((ext_vector_type(4)));

__device__ __forceinline__ unsigned nms_fkey(float f) {
    unsigned u = __float_as_uint(f);
    return (u & 0x80000000u) ? ~u : (u | 0x80000000u);
}
__device__ __forceinline__ float nms_funkey(unsigned k) {
    unsigned u = (k & 0x80000000u) ? (k ^ 0x80000000u) : ~k;
    return __uint_as_float(u);
}

__global__ void nms_init(unsigned* cnt, int planes) {
    int i = blockIdx.x * blockDim.x + threadIdx.x;
    if (i < planes) cnt[i] = 0u;
}

// Phase 1: per (plane, band) — TDM-stage full-width row bands into LDS,
// separable 7x7 max, append peak keys to per-plane candidate lists.
__global__ __launch_bounds__(1024) void nms_phase1(
    const float* __restrict__ in, unsigned* __restrict__ cnt,
    unsigned long long* __restrict__ keys, int cap, int bandsPerPlane)
{
    const int tid   = threadIdx.x;          // one thread per column
    const int plane = blockIdx.x / bandsPerPlane;
    const int band  = blockIdx.x % bandsPerPlane;
    const int y0    = band * NMS_RB;
    const int ylo   = (y0 - 3 > 0) ? (y0 - 3) : 0;
    const int yhiT  = y0 + NMS_RB + 3;
    const int yhi   = (yhiT < NMS_H) ? yhiT : NMS_H;
    const int nrows = yhi - ylo;

    __shared__ float tile[NMS_MAXROWS * NMS_W];

    const float* rowBase = in + (size_t)plane * NMS_H * NMS_W + (size_t)ylo * NMS_W;

#if __has_builtin(__builtin_amdgcn_tensor_load_to_lds) && __has_builtin(__builtin_amdgcn_s_wait_tensorcnt)
    if (tid < 32) {  // wave 0 issues the DMA; EXEC is ignored by TDM
        unsigned long long gaddr = (unsigned long long)(const void*)rowBase;
        unsigned lds_off = (unsigned)(unsigned long long)(void*)&tile[0];

        v4u g0;
        g0[0] = 1u;                                             // count=1, user desc
        g0[1] = lds_off;                                        // lds_addr
        g0[2] = (unsigned)(gaddr & 0xFFFFFFFFull);              // global_addr lo
        g0[3] = (unsigned)((gaddr >> 32) & 0x01FFFFFFull)
              | (2u << 30);                                     // global_addr hi | type=2

        v8i g1;
        g1[0] = (int)(2u << 16);                 // data_size = 4B
        g1[1] = (int)(NMS_W << 16);              // tensor_dim0[15:0] @ bits 63:48
        g1[2] = (int)(NMS_H << 16);              // tensor_dim1[15:0] @ bits 95:80
        g1[3] = (int)(NMS_W << 16);              // tile_dim0 @ bits 127:112
        g1[4] = nrows;                           // tile_dim1 (tile_dim2 = 0)
        g1[5] = NMS_W;                           // tensor_dim0_stride lo32
        g1[6] = 0;
        g1[7] = 0;

        v4i g2 = {0, 0, 0, 0};
        v4i g3 = {0, 0, 0, 0};
#if defined(__clang_major__) && (__clang_major__ >= 23)
        v8i g4 = {0, 0, 0, 0, 0, 0, 0, 0};
        __builtin_amdgcn_tensor_load_to_lds(g0, g1, g2, g3, g4, 0);
#else
        __builtin_amdgcn_tensor_load_to_lds(g0, g1, g2, g3, 0);
#endif
        __builtin_amdgcn_s_wait_tensorcnt(0);
    }
#else
    for (int i = tid; i < nrows * NMS_W; i += 1024) tile[i] = rowBase[i];
#endif
    __syncthreads();

    const int x = tid;
    // Stage A: horizontal 7-max per loaded row (clamped window), into registers.
    float r[NMS_MAXROWS];
#pragma unroll
    for (int j = 0; j < NMS_MAXROWS; ++j) {
        float m = -__builtin_inff();
        if (j < nrows) {
            const float* row = &tile[j * NMS_W];
#pragma unroll
            for (int dx = -3; dx <= 3; ++dx) {
                int xx = x + dx;
                bool ok = (xx >= 0) && (xx < NMS_W);
                int xc = xx < 0 ? 0 : (xx > NMS_W - 1 ? NMS_W - 1 : xx);
                float t = row[xc];
                m = fmaxf(m, ok ? t : -__builtin_inff());
            }
        }
        r[j] = m;
    }

    // Stage B: vertical 7-max + peak test + candidate append.
    const float thr = 2.0f / (float)(NMS_H * NMS_W);
#pragma unroll
    for (int i = 0; i < NMS_RB; ++i) {
        int y = y0 + i;
        float v = tile[(y - ylo) * NMS_W + x];
        int jlo = ((y - 3 > 0) ? (y - 3) : 0) - ylo;
        int jhiT = y + 4;
        int jhi = ((jhiT < NMS_H) ? jhiT : NMS_H) - ylo;
        float wm = -__builtin_inff();
#pragma unroll
        for (int j = 0; j < NMS_MAXROWS; ++j) {
            bool use = (j >= jlo) && (j < jhi);
            wm = fmaxf(wm, use ? r[j] : -__builtin_inff());
        }
        if (v == wm && v > thr) {
            unsigned idx = (unsigned)(y * NMS_W + x);
            unsigned long long key =
                ((unsigned long long)nms_fkey(v) << 32) | (unsigned)(~idx);
            unsigned pos = atomicAdd(&cnt[plane], 1u);
            if (pos < (unsigned)cap)
                keys[(size_t)plane * (size_t)cap + pos] = key;
        }
    }
}

// Phase 2: per-plane exact top-K via byte-wise radix select on unique 64-bit
// keys (8 histogram passes + 1 gather), then O(K^2) rank to order finalists.
__global__ __launch_bounds__(256) void nms_phase2(
    const unsigned* __restrict__ cnt, const unsigned long long* __restrict__ keys,
    int cap, int* __restrict__ skelOut, float* __restrict__ scoreOut)
{
    const int plane = blockIdx.x;
    const int tid   = threadIdx.x;
    unsigned c = cnt[plane];
    const int n = (c < (unsigned)cap) ? (int)c : cap;
    const int K = (n < NMS_K) ? n : NMS_K;
    const unsigned long long* k = keys + (size_t)plane * (size_t)cap;
    const int seg = plane % NMS_S;

    __shared__ unsigned hist[256];
    __shared__ unsigned long long sPrefix;
    __shared__ int sRemaining;
    __shared__ unsigned long long fin[128];
    __shared__ unsigned fcount;

    unsigned long long kth = 0ull;
    if (K > 0) {
        unsigned long long prefix = 0ull;
        int remaining = K;
#pragma unroll 1
        for (int d = 7; d >= 0; --d) {
            hist[tid] = 0u;
            __syncthreads();
            const int shift = d * 8;
            // mask of bytes strictly above byte d (d==7 -> no constraint)
            const unsigned long long prefMask =
                (d == 7) ? 0ull : (~0ull << (8 * (d + 1)));
            for (int i = tid; i < n; i += 256) {
                unsigned long long v = k[i];
                if ((v & prefMask) == prefix)
                    atomicAdd(&hist[(unsigned)((v >> shift) & 0xFFull)], 1u);
            }
            __syncthreads();
            if (tid == 0) {
                int acc = 0;
                int digit = 0;
                for (int b = 255; b >= 0; --b) {
                    int cb = (int)hist[b];
                    if (acc + cb >= remaining) { digit = b; break; }
                    acc += cb;
                }
                sRemaining = remaining - acc;
                sPrefix = prefix | ((unsigned long long)digit << shift);
            }
            __syncthreads();
            prefix = sPrefix;
            remaining = sRemaining;
        }
        kth = prefix;   // exact K-th largest key (keys are unique)
    }

    // Gather the K finalists (keys >= kth); order in fin[] is nondeterministic
    // but ranks below are value-based, so output is deterministic.
    if (tid == 0) fcount = 0u;
    __syncthreads();
    if (K > 0) {
        for (int i = tid; i < n; i += 256) {
            unsigned long long v = k[i];
            if (v >= kth) {
                unsigned p = atomicAdd(&fcount, 1u);
                if (p < 128u) fin[p] = v;
            }
        }
    }
    __syncthreads();

    const int nf = (int)fcount < K ? (int)fcount : K;
    if (tid < nf) {
        unsigned long long mine = fin[tid];
        int rank = 0;
        for (int m = 0; m < nf; ++m) rank += (fin[m] > mine) ? 1 : 0;
        unsigned idx = ~(unsigned)(mine & 0xFFFFFFFFull);
        int base3 = (plane * NMS_K + rank) * 3;
        skelOut[base3 + 0] = seg;
        skelOut[base3 + 1] = (int)(idx % NMS_W);   // x = w
        skelOut[base3 + 2] = (int)(idx / NMS_W);   // y = h
        scoreOut[plane * NMS_K + rank] = nms_funkey((unsigned)(mine >> 32));
    }
    // Fill unused slots [nf, K) .. 99 with -inf / zeros.
    if (tid >= nf && tid < NMS_K) {
        int base3 = (plane * NMS_K + tid) * 3;
        skelOut[base3 + 0] = seg;
        skelOut[base3 + 1] = 0;
        skelOut[base3 + 2] = 0;
        scoreOut[plane * NMS_K + tid] = -__builtin_inff();
    }
}

extern "C" void kernel_launch(void* const* d_in, const int* in_sizes, int n_in,
                              void* d_out, int out_size, void* d_ws, size_t ws_size,
                              hipStream_t stream) {
    (void)n_in; (void)out_size;
    const float* belive = (const float*)d_in[0];
    const int planes = in_sizes[0] / (NMS_H * NMS_W);   // B*S = 100

    // Workspace: [0,512): per-plane counters; then per-plane key lists.
    unsigned* cnt = (unsigned*)d_ws;
    unsigned long long* keys = (unsigned long long*)((char*)d_ws + 512);
    long long avail = ((long long)ws_size - 512) / ((long long)planes * 8);
    int cap = (int)(avail < 32768 ? avail : 32768);
    if (cap < 256) cap = 256;

    // d_out: [B,S,K,3] int32 skeletons then [B,S,K] float scores, flat.
    int*   skelOut  = (int*)d_out;
    float* scoreOut = (float*)d_out + (size_t)planes * NMS_K * 3;

    nms_init<<<(planes + 127) / 128, 128, 0, stream>>>(cnt, planes);

    const int bandsPerPlane = NMS_H / NMS_RB;           // 128
    nms_phase1<<<planes * bandsPerPlane, 1024, 0, stream>>>(
        belive, cnt, keys, cap, bandsPerPlane);

    nms_phase2<<<planes, 256, 0, stream>>>(cnt, keys, cap, skelOut, scoreOut);
}